// SpatialTemporalGNN_1580547975260
// MI455X (gfx1250) — compile-verified
//
#include <hip/hip_runtime.h>
#include <hip/hip_bf16.h>

// ---------------------------------------------------------------------------
// SpatialTemporalGNN forward for MI455X (gfx1250, wave32).
//   Phase A: GCN degree/norm precompute
//   Phase B: layer-1 rank-1 linear, self-loop init, edge scatter (atomics)
//   Phase C: layer-2 GEMM via V_WMMA_F32_16X16X4_F32, B staged in LDS via
//            GLOBAL_LOAD_ASYNC_TO_LDS_B128 (ASYNCcnt) when available
//   Phase D: init+scatter for layer 2
//   Phase E: mean pool, LSTM (single block, LDS state), output head
// ---------------------------------------------------------------------------

#define N_NODES 150000
#define N_EDGES 1500000
#define HID     100
#define NPAD    112          // HID padded to 7 WMMA N-tiles
#define N_GRAPHS 150
#define NODES_PER_GRAPH 1000
#define BATCH   30
#define SEQ     5

typedef __attribute__((ext_vector_type(2))) float v2f;
typedef __attribute__((ext_vector_type(8))) float v8f;
typedef int v4i_ __attribute__((vector_size(16)));   // matches builtin param type

#if defined(__gfx1250__) && \
    __has_builtin(__builtin_amdgcn_global_load_async_to_lds_b128) && \
    __has_builtin(__builtin_amdgcn_s_wait_asynccnt)
#define USE_ASYNC_LDS 1
#else
#define USE_ASYNC_LDS 0
#endif

// ---------------- Phase A: degree / norm ----------------

__global__ void init_deg_kernel(float* deg) {
    int n = blockIdx.x * blockDim.x + threadIdx.x;
    if (n < N_NODES) deg[n] = 1.0f;               // self loop
}

__global__ void degree_kernel(float* deg, const int* dst) {
    int e = blockIdx.x * blockDim.x + threadIdx.x;
    if (e < N_EDGES) atomicAdd(&deg[dst[e]], 1.0f);
}

__global__ void dinv_kernel(float* dinv, const float* deg) {
    int n = blockIdx.x * blockDim.x + threadIdx.x;
    if (n < N_NODES) dinv[n] = rsqrtf(deg[n]);
}

__global__ void norm_kernel(float* norm, const float* dinv, const int* src, const int* dst) {
    int e = blockIdx.x * blockDim.x + threadIdx.x;
    if (e < N_EDGES) norm[e] = dinv[src[e]] * dinv[dst[e]];
}

// ---------------- Phase B: layer-1 linear (rank-1), init, scatter ----------------

// h1[n][f] = x[n] * W1[f]       (IN_F == 1)
__global__ void h1_kernel(float* h1, const float* x, const float* W1) {
    long long i = (long long)blockIdx.x * blockDim.x + threadIdx.x;
    if (i >= (long long)N_NODES * HID) return;
    int n = (int)(i / HID);
    int f = (int)(i % HID);
    h1[i] = x[n] * W1[f];
}

// out[n][f] = b[f] + h[n][f] * dinv[n]^2    (bias + normalized self loop)
__global__ void init_out_kernel(float* out, const float* h, const float* dinv, const float* bias) {
    long long i = (long long)blockIdx.x * blockDim.x + threadIdx.x;
    if (i >= (long long)N_NODES * HID) return;
    int n = (int)(i / HID);
    int f = (int)(i % HID);
    float dv = dinv[n];
    out[i] = bias[f] + h[i] * dv * dv;
}

// out[dst[e]][f] += h[src[e]][f] * norm[e]   (hot loop: L2-resident atomics)
__global__ void scatter_kernel(float* out, const float* __restrict__ h,
                               const float* __restrict__ norm,
                               const int* __restrict__ src, const int* __restrict__ dst) {
    long long i = (long long)blockIdx.x * blockDim.x + threadIdx.x;
    if (i >= (long long)N_EDGES * HID) return;
    int e = (int)(i / HID);
    int f = (int)(i % HID);
    int s = src[e];
    int d = dst[e];
    float v = h[(long long)s * HID + f] * norm[e];
    atomicAdd(&out[(long long)d * HID + f], v);
}

// ---------------- Phase C: layer-2 GEMM via WMMA f32 16x16x4 ----------------

// W2 [100 x 100] row-major (in, out) -> padded [100 x 112]
__global__ void padW2_kernel(float* W2p, const float* W2) {
    int i = blockIdx.x * blockDim.x + threadIdx.x;
    if (i >= HID * NPAD) return;
    int k = i / NPAD;
    int n = i % NPAD;
    W2p[i] = (n < HID) ? W2[k * HID + n] : 0.0f;
}

// D[150000 x 100] = A[150000 x 100] @ Bp[100 x 112], one wave per 16x16 tile.
// Entire Bp (44.8 KB) is staged once per block into LDS (async copy when the
// toolchain exposes the gfx1250 builtins), then every wave's B-fragments are
// ds_load broadcasts instead of per-lane global loads.
//
// A-frag (16x4 f32, 2 VGPR): lane l holds A[m=l&15][k = 2*(l>>4)+j], j=0,1
// B-frag (4x16 f32, 2 VGPR): lane l holds B[k = 2*(l>>4)+j][n=l&15]
// C/D    (16x16 f32, 8 VGPR): vgpr r, lane l -> D[m = r + 8*(l>>4)][n = l&15]
__global__ void __launch_bounds__(256)
wmma_gemm_kernel(float* __restrict__ D, const float* __restrict__ A,
                 const float* __restrict__ Bp) {
    __shared__ __align__(16) float ldsB[HID * NPAD];   // 44.8 KB of 320 KB/WGP

    // ---- cooperative stage of B into LDS (all waves participate) ----
#if USE_ASYNC_LDS
    for (int i = threadIdx.x * 4; i < HID * NPAD; i += blockDim.x * 4) {
        __builtin_amdgcn_global_load_async_to_lds_b128(
            (__attribute__((address_space(1))) v4i_*)(Bp + i),
            (__attribute__((address_space(3))) v4i_*)(ldsB + i),
            /*offset=*/0, /*cpol=*/0);
    }
    __builtin_amdgcn_s_wait_asynccnt(0);
#else
    for (int i = threadIdx.x; i < HID * NPAD; i += blockDim.x) ldsB[i] = Bp[i];
#endif
    __syncthreads();

    const int NT = NPAD / 16;                 // 7 N-tiles
    const int MT = N_NODES / 16;              // 9375 M-tiles
    int wave = blockIdx.x * (blockDim.x >> 5) + (threadIdx.x >> 5);
    if (wave >= MT * NT) return;              // wave-uniform exit, EXEC stays full
    int lane  = threadIdx.x & 31;
    int mt    = wave / NT;
    int nt    = wave % NT;
    int mbase = mt * 16;
    int nbase = nt * 16;
    int l15   = lane & 15;
    int khalf = lane >> 4;                    // 0 or 1

    v8f acc = {};
    const float* arow = A + (long long)(mbase + l15) * HID + 2 * khalf;
    const float* brow = ldsB + nbase + l15 + 2 * khalf * NPAD;
    #pragma unroll 5
    for (int kk = 0; kk < HID / 4; ++kk) {
        v2f a;
        a.x = arow[kk * 4 + 0];
        a.y = arow[kk * 4 + 1];
        v2f b;
        b.x = brow[kk * 4 * NPAD];
        b.y = brow[kk * 4 * NPAD + NPAD];
        acc = __builtin_amdgcn_wmma_f32_16x16x4_f32(
            /*neg_a=*/false, a, /*neg_b=*/false, b,
            /*c_mod=*/(short)0, acc, /*reuse_a=*/false, /*reuse_b=*/false);
    }

    int col = nbase + l15;
    if (col < HID) {
        #pragma unroll
        for (int r = 0; r < 8; ++r) {
            int row = mbase + r + 8 * khalf;
            D[(long long)row * HID + col] = acc[r];
        }
    }
}

// ---------------- Phase E: pool + LSTM + head ----------------

// pooled[g][f] = mean over 1000 contiguous nodes of out2
__global__ void pool_kernel(float* pooled, const float* __restrict__ out2) {
    int i = blockIdx.x * blockDim.x + threadIdx.x;
    if (i >= N_GRAPHS * HID) return;
    int g = i / HID;
    int f = i % HID;
    const float* base = out2 + (long long)g * NODES_PER_GRAPH * HID + f;
    float s = 0.0f;
    for (int n = 0; n < NODES_PER_GRAPH; ++n) s += base[(long long)n * HID];
    pooled[i] = s * (1.0f / (float)NODES_PER_GRAPH);
}

__device__ __forceinline__ float sigmoidf_(float x) { return 1.0f / (1.0f + __expf(-x)); }

// Single block; LDS-resident h/c state; PyTorch gate order i,f,g,o.
__global__ void __launch_bounds__(1024)
lstm_head_kernel(float* __restrict__ out, const float* __restrict__ pooled,
                 const float* __restrict__ W_ih, const float* __restrict__ W_hh,
                 const float* __restrict__ b_ih, const float* __restrict__ b_hh,
                 const float* __restrict__ W_out, const float* __restrict__ b_out) {
    __shared__ float hS[BATCH * HID];
    __shared__ float hN[BATCH * HID];
    __shared__ float cS[BATCH * HID];
    int tid = threadIdx.x;
    for (int i = tid; i < BATCH * HID; i += blockDim.x) { hS[i] = 0.0f; cS[i] = 0.0f; }
    __syncthreads();

    for (int t = 0; t < SEQ; ++t) {
        for (int idx = tid; idx < BATCH * HID; idx += blockDim.x) {
            int b = idx / HID;
            int u = idx % HID;
            const float* xt = pooled + (b * SEQ + t) * HID;
            const float* hb = hS + b * HID;
            float g[4];
            #pragma unroll
            for (int gi = 0; gi < 4; ++gi) {
                int row = gi * HID + u;
                const float* wi = W_ih + row * HID;
                const float* wh = W_hh + row * HID;
                float acc = b_ih[row] + b_hh[row];
                for (int k = 0; k < HID; ++k) acc += xt[k] * wi[k] + hb[k] * wh[k];
                g[gi] = acc;
            }
            float cn = sigmoidf_(g[1]) * cS[idx] + sigmoidf_(g[0]) * tanhf(g[2]);
            cS[idx] = cn;
            hN[idx] = sigmoidf_(g[3]) * tanhf(cn);
        }
        __syncthreads();
        for (int i = tid; i < BATCH * HID; i += blockDim.x) hS[i] = hN[i];
        __syncthreads();
    }

    if (tid < BATCH) {
        float acc = b_out[0];
        const float* hb = hS + tid * HID;
        for (int k = 0; k < HID; ++k) acc += hb[k] * W_out[k];
        out[tid] = acc;
    }
}

// ---------------------------------------------------------------------------

static inline int cdiv(long long a, int b) { return (int)((a + b - 1) / b); }

extern "C" void kernel_launch(void* const* d_in, const int* in_sizes, int n_in,
                              void* d_out, int out_size, void* d_ws, size_t ws_size,
                              hipStream_t stream) {
    const float* x     = (const float*)d_in[0];
    const int*   eidx  = (const int*)d_in[1];
    // d_in[2] = batch (contiguous 1000-node graphs; layout known, unused)
    const float* W1    = (const float*)d_in[3];
    const float* b1    = (const float*)d_in[4];
    const float* W2    = (const float*)d_in[5];
    const float* b2    = (const float*)d_in[6];
    const float* W_ih  = (const float*)d_in[7];
    const float* W_hh  = (const float*)d_in[8];
    const float* b_ih  = (const float*)d_in[9];
    const float* b_hh  = (const float*)d_in[10];
    const float* W_out = (const float*)d_in[11];
    const float* b_out = (const float*)d_in[12];
    float* out = (float*)d_out;

    const int* src = eidx;
    const int* dst = eidx + N_EDGES;

    // Workspace carve-up (bytes)
    char* ws = (char*)d_ws;
    const size_t FEAT_BYTES = (size_t)N_NODES * HID * sizeof(float);   // 60 MB
    float* bufA   = (float*)(ws);                                      // h1 / h2lin
    float* bufB   = (float*)(ws + FEAT_BYTES);                         // out1 / out2
    char*  small  = ws + 2 * FEAT_BYTES;
    float* deg    = (float*)(small);                 small += (size_t)N_NODES * 4;
    float* dinv   = (float*)(small);                 small += (size_t)N_NODES * 4;
    float* normE  = (float*)(small);                 small += (size_t)N_EDGES * 4;
    float* W2p    = (float*)(small);                 small += (size_t)HID * NPAD * 4;
    float* pooled = (float*)(small);                 small += (size_t)N_GRAPHS * HID * 4;

    const int B256 = 256;
    const long long NF = (long long)N_NODES * HID;
    const long long EF = (long long)N_EDGES * HID;
    const int gemm_waves = (N_NODES / 16) * (NPAD / 16);     // 65625 tiles

    // Phase A: degrees & norms
    init_deg_kernel<<<cdiv(N_NODES, B256), B256, 0, stream>>>(deg);
    degree_kernel<<<cdiv(N_EDGES, B256), B256, 0, stream>>>(deg, dst);
    dinv_kernel<<<cdiv(N_NODES, B256), B256, 0, stream>>>(dinv, deg);
    norm_kernel<<<cdiv(N_EDGES, B256), B256, 0, stream>>>(normE, dinv, src, dst);

    // Phase B: layer-1 linear + propagate
    h1_kernel<<<cdiv(NF, B256), B256, 0, stream>>>(bufA, x, W1);
    init_out_kernel<<<cdiv(NF, B256), B256, 0, stream>>>(bufB, bufA, dinv, b1);
    scatter_kernel<<<cdiv(EF, B256), B256, 0, stream>>>(bufB, bufA, normE, src, dst);

    // Phase C: layer-2 linear via WMMA (f32-exact), LDS-staged B
    padW2_kernel<<<cdiv(HID * NPAD, B256), B256, 0, stream>>>(W2p, W2);
    wmma_gemm_kernel<<<cdiv(gemm_waves, B256 / 32), B256, 0, stream>>>(bufA, bufB, W2p);

    // Phase D: layer-2 propagate
    init_out_kernel<<<cdiv(NF, B256), B256, 0, stream>>>(bufB, bufA, dinv, b2);
    scatter_kernel<<<cdiv(EF, B256), B256, 0, stream>>>(bufB, bufA, normE, src, dst);

    // Phase E: pool + LSTM + head
    pool_kernel<<<cdiv(N_GRAPHS * HID, B256), B256, 0, stream>>>(pooled, bufB);
    lstm_head_kernel<<<1, 1024, 0, stream>>>(out, pooled, W_ih, W_hh, b_ih, b_hh, W_out, b_out);

    (void)in_sizes; (void)n_in; (void)out_size; (void)ws_size;
}